// Protonet_55387898250096
// MI455X (gfx1250) — compile-verified
//
#include <hip/hip_runtime.h>
#include <math.h>

// ---------------- problem constants ----------------
#define V_EMB   128
#define BDIM    4096
#define CDIM    50
#define MSIZ    200000
#define FOCN    512
#define DKF     11.313708498984761f   // sqrt(128)
#define EPS_COS 1e-8f
#define EPS_NRM 1e-12f

// ---------------- ws layout (float offsets) ----------------
#define OFF_SIM    0u         // 4096*50
#define OFF_SCORE  204800u    // 4096*50
#define OFF_ATT2   409600u    // 4096*128
#define OFF_PROTO  933888u    // 128
#define OFF_T0     934016u    // 128
#define OFF_SIMM   934144u    // 200000
#define OFF_WPOS   1134144u   // 200000
#define OFF_WNEG   1334144u   // 200000
#define OFF_SCAL   1534144u   // 64 scalars: 0=maxp 1=minsim 2=denp 3=denn 6=|t0|
#define OFF_PMAX   1534208u   // 3125 (pad 4096)
#define OFF_PMIN   1538304u   // 3125 (pad 4096)
#define OFF_PSP    1542400u   // 400 (pad 1024)
#define OFF_PSN    1543424u   // 400 (pad 1024)
#define OFF_PP     1544448u   // 32*128
#define OFF_MPART  1548544u   // 200*256
#define OFF_MEMC   1599744u   // 128
#define OFF_NMEMC  1599872u   // 128
#define OFF_COSB   1600000u   // 4096
#define OFF_UINTS  1604096u   // uint region: hist[256], 256=pfx_hi 257=pfx_lo 258=remaining

typedef __attribute__((ext_vector_type(2))) float v2f;
typedef __attribute__((ext_vector_type(8))) float v8f;

__device__ __forceinline__ float wave_sum(float v) {
  for (int o = 16; o; o >>= 1) v += __shfl_xor(v, o, 32);
  return v;
}
__device__ __forceinline__ unsigned f2mono(float f) {
  unsigned u = __float_as_uint(f);
  return (u & 0x80000000u) ? ~u : (u | 0x80000000u);
}
__device__ __forceinline__ float logsig(float x) {
  return fminf(x, 0.f) - log1pf(expf(-fabsf(x)));
}

// t0 = W_target[target[0]]; scal[6] = ||t0||
__global__ void k_t0(const int* __restrict__ target, const float* __restrict__ Wt,
                     float* __restrict__ t0, float* __restrict__ scal) {
  __shared__ float red[V_EMB];
  int e = threadIdx.x;
  float v = Wt[(size_t)target[0] * V_EMB + e];
  t0[e] = v;
  red[e] = v * v; __syncthreads();
  for (int s = 64; s; s >>= 1) { if (e < s) red[e] += red[e + s]; __syncthreads(); }
  if (e == 0) scal[6] = sqrtf(red[0]);
}

// sim[b,c] = dot(W_target[target[b]], W_contex[contex[b,c]])
__global__ void k_sim(const int* __restrict__ target, const int* __restrict__ contex,
                      const float* __restrict__ Wt, const float* __restrict__ Wc,
                      float* __restrict__ sim) {
  __shared__ float tgt[V_EMB];
  int b = blockIdx.x, tid = threadIdx.x;
  int ti = target[b];
  if (tid < V_EMB) tgt[tid] = Wt[(size_t)ti * V_EMB + tid];
  __syncthreads();
  int wave = tid >> 5, lane = tid & 31;
  for (int c = wave; c < CDIM; c += 8) {
    const float* row = Wc + (size_t)contex[b * CDIM + c] * V_EMB;
    float acc = 0.f;
    for (int j = lane; j < V_EMB; j += 32) acc += row[j] * tgt[j];
    acc = wave_sum(acc);
    if (lane == 0) sim[b * CDIM + c] = acc;
  }
}

// score[b,c] = softmax over b (axis 0!) of sim[:,c], divided by sqrt(E)
__global__ void k_softmax0(const float* __restrict__ sim, float* __restrict__ score) {
  __shared__ float red[256];
  int c = blockIdx.x, tid = threadIdx.x;
  float m = -__builtin_inff();
  for (int b = tid; b < BDIM; b += 256) m = fmaxf(m, sim[b * CDIM + c]);
  red[tid] = m; __syncthreads();
  for (int s = 128; s; s >>= 1) { if (tid < s) red[tid] = fmaxf(red[tid], red[tid + s]); __syncthreads(); }
  float cm = red[0]; __syncthreads();
  float sum = 0.f;
  for (int b = tid; b < BDIM; b += 256) sum += expf(sim[b * CDIM + c] - cm);
  red[tid] = sum; __syncthreads();
  for (int s = 128; s; s >>= 1) { if (tid < s) red[tid] += red[tid + s]; __syncthreads(); }
  float inv = 1.f / (red[0] * DKF);
  for (int b = tid; b < BDIM; b += 256)
    score[b * CDIM + c] = expf(sim[b * CDIM + c] - cm) * inv;
}

// att2[b,e] = sum_c score[b,c] * W_contex[contex[b,c]][e]
__global__ void k_attend(const int* __restrict__ contex, const float* __restrict__ Wc,
                         const float* __restrict__ score, float* __restrict__ att2) {
  __shared__ float sc[CDIM];
  __shared__ int ci[CDIM];
  int b = blockIdx.x, e = threadIdx.x;
  if (e < CDIM) { sc[e] = score[b * CDIM + e]; ci[e] = contex[b * CDIM + e]; }
  __syncthreads();
  float acc = 0.f;
  for (int c = 0; c < CDIM; ++c) acc += sc[c] * Wc[(size_t)ci[c] * V_EMB + e];
  att2[(size_t)b * V_EMB + e] = acc;
}

__global__ void k_proto_partial(const float* __restrict__ att2, float* __restrict__ pp) {
  int j = blockIdx.x, e = threadIdx.x;
  float acc = 0.f;
  for (int b = j * 128; b < j * 128 + 128; ++b) acc += att2[(size_t)b * V_EMB + e];
  pp[j * V_EMB + e] = acc;
}
__global__ void k_proto_reduce(const float* __restrict__ pp, float* __restrict__ proto) {
  int e = threadIdx.x;
  float acc = 0.f;
  for (int j = 0; j < 32; ++j) acc += pp[j * V_EMB + e];
  proto[e] = acc * (1.f / (float)BDIM);
}

// ---- WMMA pass: sim_m[m] = dot(proto_contex, proto_memory[m]) ----
// One wave = one 16-row tile. A (16x4 f32) is proto_contex replicated across
// all 16 M-rows, B (4x16 f32) columns are 16 memory rows; K-loop of 32 WMMAs.
// 3125 blocks x 4 waves x 16 rows = 200000 exactly.
__global__ void k_simm(const float* __restrict__ mem, const float* __restrict__ pc,
                       float* __restrict__ simm, float* __restrict__ pmax,
                       float* __restrict__ pmin) {
  __shared__ float smax[4], smin[4];
  int tid = threadIdx.x, wave = tid >> 5, lane = tid & 31;
  size_t row0 = ((size_t)blockIdx.x * 4 + wave) * 16;
  int ks = (lane < 16) ? 0 : 2;                    // K-pair selected by lane half
  const float* rowp = mem + (row0 + (size_t)(lane & 15)) * V_EMB;
  v8f c = {0.f, 0.f, 0.f, 0.f, 0.f, 0.f, 0.f, 0.f};
#pragma unroll
  for (int kk = 0; kk < V_EMB; kk += 4) {
    v2f a = *(const v2f*)(pc + kk + ks);           // A: replicated proto chunk
    v2f b = *(const v2f*)(rowp + kk + ks);         // B: memory row chunk
    c = __builtin_amdgcn_wmma_f32_16x16x4_f32(false, a, false, b, (short)0, c,
                                              false, false);
  }
  float v = c[0];                                  // M=0 (or M=8) row: all equal
  if (lane < 16) simm[row0 + lane] = v;
  float mx = v, mn = v;
  for (int o = 16; o; o >>= 1) {
    mx = fmaxf(mx, __shfl_xor(mx, o, 32));
    mn = fminf(mn, __shfl_xor(mn, o, 32));
  }
  if (lane == 0) { smax[wave] = mx; smin[wave] = mn; }
  __syncthreads();
  if (tid == 0) {
    float MX = smax[0], MN = smin[0];
    for (int w = 1; w < 4; ++w) { MX = fmaxf(MX, smax[w]); MN = fminf(MN, smin[w]); }
    pmax[blockIdx.x] = MX; pmin[blockIdx.x] = MN;
  }
}

__global__ void k_minmax(const float* __restrict__ pmax, const float* __restrict__ pmin,
                         float* __restrict__ scal) {
  __shared__ float r1[256], r2[256];
  int t = threadIdx.x;
  float mx = -__builtin_inff(), mn = __builtin_inff();
  for (int i = t; i < 3125; i += 256) { mx = fmaxf(mx, pmax[i]); mn = fminf(mn, pmin[i]); }
  r1[t] = mx; r2[t] = mn; __syncthreads();
  for (int s = 128; s; s >>= 1) {
    if (t < s) { r1[t] = fmaxf(r1[t], r1[t + s]); r2[t] = fminf(r2[t], r2[t + s]); }
    __syncthreads();
  }
  if (t == 0) { scal[0] = r1[0]; scal[1] = r2[0]; }
}

__global__ void k_expsums(const float* __restrict__ simm, const float* __restrict__ scal,
                          float* __restrict__ psp, float* __restrict__ psn) {
  __shared__ float r1[256], r2[256];
  int t = threadIdx.x;
  float maxp = scal[0], minsim = scal[1];
  float sp = 0.f, sn = 0.f;
  for (int m = blockIdx.x * 256 + t; m < MSIZ; m += 400 * 256) {
    float s = simm[m];
    sp += expf(s - maxp);
    sn += expf((minsim - s) / DKF);    // exponent = -s/DK - (-minsim/DK)
  }
  r1[t] = sp; r2[t] = sn; __syncthreads();
  for (int s = 128; s; s >>= 1) {
    if (t < s) { r1[t] += r1[t + s]; r2[t] += r2[t + s]; }
    __syncthreads();
  }
  if (t == 0) { psp[blockIdx.x] = r1[0]; psn[blockIdx.x] = r2[0]; }
}

__global__ void k_denoms(const float* __restrict__ psp, const float* __restrict__ psn,
                         float* __restrict__ scal) {
  __shared__ float r1[256], r2[256];
  int t = threadIdx.x;
  float a = 0.f, b = 0.f;
  for (int i = t; i < 400; i += 256) { a += psp[i]; b += psn[i]; }
  r1[t] = a; r2[t] = b; __syncthreads();
  for (int s = 128; s; s >>= 1) {
    if (t < s) { r1[t] += r1[t + s]; r2[t] += r2[t + s]; }
    __syncthreads();
  }
  if (t == 0) { scal[2] = r1[0]; scal[3] = r2[0]; }
}

__global__ void k_weights(const float* __restrict__ simm, const float* __restrict__ scal,
                          float* __restrict__ wpos, float* __restrict__ wneg) {
  int m = blockIdx.x * 256 + threadIdx.x;
  if (m >= MSIZ) return;
  float s = simm[m];
  wpos[m] = expf(s - scal[0]) / (scal[2] * DKF);        // softmax(sim)/DK
  wneg[m] = expf((scal[1] - s) / DKF) / scal[3];        // softmax(-sim/DK)
}

// mem_contex / neg_mem_contex partial weighted row-sums (1000 rows per block)
__global__ void k_memvec_partial(const float* __restrict__ mem, const float* __restrict__ wpos,
                                 const float* __restrict__ wneg, float* __restrict__ mpart) {
  int j = blockIdx.x, e = threadIdx.x;
  float ap = 0.f, an = 0.f;
  for (int m = j * 1000; m < j * 1000 + 1000; ++m) {
    __builtin_prefetch(mem + (size_t)(m + 8) * V_EMB + e, 0, 0);
    float x = mem[(size_t)m * V_EMB + e];
    ap += wpos[m] * x;
    an += wneg[m] * x;
  }
  mpart[j * 256 + e] = ap;
  mpart[j * 256 + 128 + e] = an;
}
__global__ void k_memvec_reduce(const float* __restrict__ mpart, float* __restrict__ memc,
                                float* __restrict__ nmemc) {
  int e = threadIdx.x;
  float a = 0.f, b = 0.f;
  for (int j = 0; j < 200; ++j) { a += mpart[j * 256 + e]; b += mpart[j * 256 + 128 + e]; }
  memc[e] = a; nmemc[e] = b;
}

// ---- radix top-512 select on 64-bit keys (value desc, index asc) ----
__global__ void k_radix_init(unsigned* __restrict__ u) {
  int t = threadIdx.x;
  if (t < 256) u[t] = 0u;
  if (t == 256) { u[256] = 0u; u[257] = 0u; u[258] = (unsigned)FOCN; }
}
__global__ void k_radix_hist(const float* __restrict__ simm, unsigned* __restrict__ u, int pass) {
  __shared__ unsigned h[256];
  int t = threadIdx.x;
  if (t < 256) h[t] = 0u;
  __syncthreads();
  unsigned long long prefix = (((unsigned long long)u[256]) << 32) | u[257];
  int shift_p = 64 - 8 * pass, shift_b = 56 - 8 * pass;
  for (int m = blockIdx.x * 256 + t; m < MSIZ; m += 400 * 256) {
    unsigned long long key =
        (((unsigned long long)f2mono(simm[m])) << 32) | (0xFFFFFFFFu - (unsigned)m);
    bool cand = (pass == 0) || ((key >> shift_p) == prefix);
    if (cand) atomicAdd(&h[(unsigned)((key >> shift_b) & 0xFFull)], 1u);
  }
  __syncthreads();
  if (t < 256 && h[t]) atomicAdd(&u[t], h[t]);
}
__global__ void k_radix_pick(unsigned* __restrict__ u) {
  if (threadIdx.x == 0) {
    unsigned rem = u[258], cum = 0u;
    int chosen = 0;
    for (int b = 255; b >= 0; --b) {
      unsigned hb = u[b];
      cum += hb;
      if (cum >= rem) { chosen = b; rem -= (cum - hb); break; }
    }
    unsigned long long prefix = (((unsigned long long)u[256]) << 32) | u[257];
    prefix = (prefix << 8) | (unsigned)chosen;
    u[256] = (unsigned)(prefix >> 32);
    u[257] = (unsigned)prefix;
    u[258] = rem;
  }
  __syncthreads();
  int t = threadIdx.x;
  if (t < 256) u[t] = 0u;   // zero hist for next pass
}

// copy memory to output; rows in top-512 set get rank-1 update + renormalize
__global__ void k_update(const float* __restrict__ mem, const float* __restrict__ simm,
                         const float* __restrict__ wpos, const float* __restrict__ proto,
                         const unsigned* __restrict__ u, const int* __restrict__ repe,
                         float* __restrict__ out) {
  __shared__ float red[V_EMB];
  int m = blockIdx.x, e = threadIdx.x;
  unsigned long long T = (((unsigned long long)u[256]) << 32) | u[257];
  unsigned long long key =
      (((unsigned long long)f2mono(simm[m])) << 32) | (0xFFFFFFFFu - (unsigned)m);
  float x = mem[(size_t)m * V_EMB + e];
  float o;
  if ((repe[0] != 0) && (key >= T)) {
    float s = x + wpos[m] * proto[e];
    red[e] = s * s; __syncthreads();
    for (int st = 64; st; st >>= 1) { if (e < st) red[e] += red[e + st]; __syncthreads(); }
    o = s / fmaxf(sqrtf(red[0]), EPS_NRM);
  } else {
    o = x;
  }
  out[1 + (size_t)m * V_EMB + e] = o;
}

__global__ void k_cos_rows(const float* __restrict__ att2, const float* __restrict__ t0,
                           const float* __restrict__ scal, float* __restrict__ cosb) {
  __shared__ float r1[V_EMB], r2[V_EMB];
  int b = blockIdx.x, e = threadIdx.x;
  float x = att2[(size_t)b * V_EMB + e], t = t0[e];
  r1[e] = x * t; r2[e] = x * x; __syncthreads();
  for (int s = 64; s; s >>= 1) {
    if (e < s) { r1[e] += r1[e + s]; r2[e] += r2[e + s]; }
    __syncthreads();
  }
  if (e == 0)
    cosb[b] = r1[0] / (fmaxf(sqrtf(r2[0]), EPS_COS) * fmaxf(scal[6], EPS_COS));
}

__global__ void k_loss(const float* __restrict__ cosb, const float* __restrict__ proto,
                       const float* __restrict__ memc, const float* __restrict__ nmemc,
                       const float* __restrict__ t0, const float* __restrict__ scal,
                       float* __restrict__ out) {
  __shared__ float r1[V_EMB], r2[V_EMB];
  int e = threadIdx.x;
  float nt = fmaxf(scal[6], EPS_COS);
  float t = t0[e];

  float s = 0.f;
  for (int b = e; b < BDIM; b += V_EMB) s += cosb[b];
  r1[e] = s; __syncthreads();
  for (int st = 64; st; st >>= 1) { if (e < st) r1[e] += r1[e + st]; __syncthreads(); }
  float S = r1[0]; __syncthreads();

  r1[e] = proto[e] * t; r2[e] = proto[e] * proto[e]; __syncthreads();
  for (int st = 64; st; st >>= 1) {
    if (e < st) { r1[e] += r1[e + st]; r2[e] += r2[e + st]; }
    __syncthreads();
  }
  float cosp = r1[0] / (fmaxf(sqrtf(r2[0]), EPS_COS) * nt); __syncthreads();

  r1[e] = memc[e] * t; r2[e] = memc[e] * memc[e]; __syncthreads();
  for (int st = 64; st; st >>= 1) {
    if (e < st) { r1[e] += r1[e + st]; r2[e] += r2[e + st]; }
    __syncthreads();
  }
  float cosm = r1[0] / (fmaxf(sqrtf(r2[0]), EPS_COS) * nt); __syncthreads();

  r1[e] = nmemc[e] * t; r2[e] = nmemc[e] * nmemc[e]; __syncthreads();
  for (int st = 64; st; st >>= 1) {
    if (e < st) { r1[e] += r1[e + st]; r2[e] += r2[e + st]; }
    __syncthreads();
  }
  float cosn = r1[0] / (fmaxf(sqrtf(r2[0]), EPS_COS) * nt);

  if (e == 0) {
    float conte = (S + cosp + cosm) / (float)(BDIM + 2);
    float negl = -cosn;
    out[0] = -(logsig(conte) + logsig(negl));
  }
}

extern "C" void kernel_launch(void* const* d_in, const int* in_sizes, int n_in,
                              void* d_out, int out_size, void* d_ws, size_t ws_size,
                              hipStream_t stream) {
  const int*   target = (const int*)d_in[0];
  const int*   contex = (const int*)d_in[1];
  const float* Wt     = (const float*)d_in[2];
  const float* Wc     = (const float*)d_in[3];
  const float* mem    = (const float*)d_in[4];
  const int*   repe   = (const int*)d_in[5];
  float* out = (float*)d_out;
  float* w   = (float*)d_ws;
  unsigned* u = (unsigned*)(w + OFF_UINTS);

  k_t0<<<1, 128, 0, stream>>>(target, Wt, w + OFF_T0, w + OFF_SCAL);
  k_sim<<<BDIM, 256, 0, stream>>>(target, contex, Wt, Wc, w + OFF_SIM);
  k_softmax0<<<CDIM, 256, 0, stream>>>(w + OFF_SIM, w + OFF_SCORE);
  k_attend<<<BDIM, 128, 0, stream>>>(contex, Wc, w + OFF_SCORE, w + OFF_ATT2);
  k_proto_partial<<<32, 128, 0, stream>>>(w + OFF_ATT2, w + OFF_PP);
  k_proto_reduce<<<1, 128, 0, stream>>>(w + OFF_PP, w + OFF_PROTO);

  k_simm<<<3125, 128, 0, stream>>>(mem, w + OFF_PROTO, w + OFF_SIMM,
                                   w + OFF_PMAX, w + OFF_PMIN);
  k_minmax<<<1, 256, 0, stream>>>(w + OFF_PMAX, w + OFF_PMIN, w + OFF_SCAL);
  k_expsums<<<400, 256, 0, stream>>>(w + OFF_SIMM, w + OFF_SCAL, w + OFF_PSP, w + OFF_PSN);
  k_denoms<<<1, 256, 0, stream>>>(w + OFF_PSP, w + OFF_PSN, w + OFF_SCAL);
  k_weights<<<(MSIZ + 255) / 256, 256, 0, stream>>>(w + OFF_SIMM, w + OFF_SCAL,
                                                    w + OFF_WPOS, w + OFF_WNEG);
  k_memvec_partial<<<200, 128, 0, stream>>>(mem, w + OFF_WPOS, w + OFF_WNEG, w + OFF_MPART);
  k_memvec_reduce<<<1, 128, 0, stream>>>(w + OFF_MPART, w + OFF_MEMC, w + OFF_NMEMC);

  k_radix_init<<<1, 512, 0, stream>>>(u);
  for (int p = 0; p < 8; ++p) {
    k_radix_hist<<<400, 256, 0, stream>>>(w + OFF_SIMM, u, p);
    k_radix_pick<<<1, 256, 0, stream>>>(u);
  }

  k_update<<<MSIZ, 128, 0, stream>>>(mem, w + OFF_SIMM, w + OFF_WPOS, w + OFF_PROTO,
                                     u, repe, out);
  k_cos_rows<<<BDIM, 128, 0, stream>>>(w + OFF_ATT2, w + OFF_T0, w + OFF_SCAL, w + OFF_COSB);
  k_loss<<<1, 128, 0, stream>>>(w + OFF_COSB, w + OFF_PROTO, w + OFF_MEMC,
                                w + OFF_NMEMC, w + OFF_T0, w + OFF_SCAL, out);
}